// UnitaryLayer_64819646431559
// MI455X (gfx1250) — compile-verified
//
#include <hip/hip_runtime.h>

// Problem constants (fixed by the reference).
constexpr int kB  = 32768;  // batch
constexpr int kD  = 1024;   // feature dim
constexpr int kNV = 128;    // number of Householder vectors

typedef __attribute__((ext_vector_type(2))) float v2f;
typedef __attribute__((ext_vector_type(8))) float v8f;

// ---------------------------------------------------------------------------
// Kernel 1: per-column scale s[k] = 2/||v[:,k]||^2 and transpose v -> vt[k][d]
// ---------------------------------------------------------------------------
__global__ __launch_bounds__(256) void uk_prep(const float* __restrict__ v,
                                               float* __restrict__ s,
                                               float* __restrict__ vt) {
  const int k = blockIdx.x;
  const int tid = threadIdx.x;
  __shared__ float red[256];
  float p = 0.f;
  for (int j = tid; j < kD; j += 256) {
    float val = v[j * kNV + k];           // column k of v (D x NV row-major)
    vt[k * kD + j] = val;                 // contiguous row for later reuse
    p += val * val;
  }
  red[tid] = p;
  __syncthreads();
  for (int off = 128; off > 0; off >>= 1) {
    if (tid < off) red[tid] += red[tid + off];
    __syncthreads();
  }
  if (tid == 0) s[k] = 2.0f / red[0];
}

// ---------------------------------------------------------------------------
// Kernel 2: build Qt = (H_0 H_1 ... H_{NV-1})^T row chains (rows independent).
// Each wave owns 4 rows of Q in registers; the result is written TRANSPOSED
// (Qt[d][r] = Q[r][d]) so the GEMM's B fragments are contiguous along K.
// ---------------------------------------------------------------------------
__global__ __launch_bounds__(256) void uk_house(const float* __restrict__ vt,
                                                const float* __restrict__ s,
                                                float* __restrict__ Qt) {
  const int lane = threadIdx.x & 31;
  const int wave = threadIdx.x >> 5;
  const int rbase = (blockIdx.x * 8 + wave) * 4;

  float q[4][32];
#pragma unroll
  for (int i = 0; i < 4; ++i)
#pragma unroll
    for (int j = 0; j < 32; ++j)
      q[i][j] = (j * 32 + lane == rbase + i) ? 1.0f : 0.0f;

  for (int k = 0; k < kNV; ++k) {
    float vv[32];
#pragma unroll
    for (int j = 0; j < 32; ++j) vv[j] = vt[k * kD + j * 32 + lane];

    float d0 = 0.f, d1 = 0.f, d2 = 0.f, d3 = 0.f;
#pragma unroll
    for (int j = 0; j < 32; ++j) {
      d0 += q[0][j] * vv[j];
      d1 += q[1][j] * vv[j];
      d2 += q[2][j] * vv[j];
      d3 += q[3][j] * vv[j];
    }
#pragma unroll
    for (int off = 16; off > 0; off >>= 1) {
      d0 += __shfl_xor(d0, off, 32);
      d1 += __shfl_xor(d1, off, 32);
      d2 += __shfl_xor(d2, off, 32);
      d3 += __shfl_xor(d3, off, 32);
    }
    const float sk = s[k];
    const float c0 = sk * d0, c1 = sk * d1, c2 = sk * d2, c3 = sk * d3;
#pragma unroll
    for (int j = 0; j < 32; ++j) {
      q[0][j] -= c0 * vv[j];
      q[1][j] -= c1 * vv[j];
      q[2][j] -= c2 * vv[j];
      q[3][j] -= c3 * vv[j];
    }
  }
  // transposed store: Qt[d][r] = Q[r][d]; 16B per wave per d, L2 absorbs it
#pragma unroll
  for (int j = 0; j < 32; ++j) {
    const int d = j * 32 + lane;
#pragma unroll
    for (int i = 0; i < 4; ++i) Qt[(size_t)d * kD + rbase + i] = q[i][j];
  }
}

// ---------------------------------------------------------------------------
// Kernel 3: out = x @ Q + b with V_WMMA_F32_16X16X4_F32 and double-buffered
// GLOBAL_LOAD_ASYNC_TO_LDS_B128 staging (ASYNCcnt pipeline).
// Block = 128 threads (4 waves, 2x2), block tile 128(M) x 64(N), K slab 32.
// Wave tile 64x32 -> 4x2 accumulators of 16x16 f32 (8 WMMAs per kk step).
// Both LDS slabs use stride 36 (K-major rows): mult of 4 -> async b128 writes
// are 16B aligned; 36*drow===2 (mod 64) unsolvable -> paired b64 fragment
// loads are bank-conflict-free. B fragments are now single ds_load_b64s
// because Q is stored transposed (contiguous along K).
// ---------------------------------------------------------------------------
constexpr int kST = 36;  // slab stride (floats) for both x and Qt tiles

__device__ __forceinline__ uint32_t lds_off(const void* p) {
  // Low 32 bits of a generic LDS pointer = workgroup-relative LDS byte addr
  return (uint32_t)(uintptr_t)p;
}

__device__ __forceinline__ void async_copy_b128(uint32_t lds_byte,
                                                const float* base,
                                                int byte_off) {
  // GVS form: global address is the 64-bit SGPR base plus the 32-bit offset in vaddr; LDS address in vdst
  asm volatile("global_load_async_to_lds_b128 %0, %1, %2"
               :
               : "v"(lds_byte), "v"(byte_off), "s"(base)
               : "memory");
}

__global__ __launch_bounds__(128) void uk_gemm(const float* __restrict__ x,
                                               const float* __restrict__ Qt,
                                               const float* __restrict__ bias,
                                               float* __restrict__ out) {
  __shared__ float sx[2][128 * kST];  // 2 x 18432 B (x rows, K-major)
  __shared__ float sq[2][64 * kST];   // 2 x  9216 B (Qt rows = N, K-major)

  const int tid = threadIdx.x;
  const int lane = tid & 31;
  const int wave = tid >> 5;
  const int wm = wave & 1;   // wave M position (0..1), 64 rows each
  const int wn = wave >> 1;  // wave N position (0..1), 32 cols each
  const int lmod = lane & 15;
  const int lhalf = lane >> 4;

  const int mBase = blockIdx.y * 128;
  const int nBase = blockIdx.x * 64;

  v8f acc[4][2];
#pragma unroll
  for (int a = 0; a < 4; ++a)
#pragma unroll
    for (int b = 0; b < 2; ++b)
      acc[a][b] = (v8f){0.f, 0.f, 0.f, 0.f, 0.f, 0.f, 0.f, 0.f};

  // ---- async staging of one K-slab into buffer `buf` -----------------------
  auto stage = [&](int buf, int k0) {
    // x tile: 128 rows x 32 K, 8 b128 per thread
#pragma unroll
    for (int i = 0; i < 8; ++i) {
      const int t = tid + i * 128;     // 0..1023
      const int m = t >> 3;            // 0..127
      const int kk4 = (t & 7) << 2;    // 0,4,...,28
      async_copy_b128(lds_off(&sx[buf][m * kST + kk4]), x,
                      (int)(((size_t)(mBase + m) * kD + k0 + kk4) * 4));
    }
    // Qt tile: 64 rows (N) x 32 K, 4 b128 per thread
#pragma unroll
    for (int i = 0; i < 4; ++i) {
      const int t = tid + i * 128;     // 0..511
      const int n = t >> 3;            // 0..63
      const int kk4 = (t & 7) << 2;    // 0,4,...,28
      async_copy_b128(lds_off(&sq[buf][n * kST + kk4]), Qt,
                      (int)(((size_t)(nBase + n) * kD + k0 + kk4) * 4));
    }
  };

  stage(0, 0);  // prologue: tile 0 in flight (12 async instrs per thread)

  for (int kb = 0; kb < kD / 32; ++kb) {
    const int p = kb & 1;
    if (kb + 1 < kD / 32) {
      stage(1 - p, (kb + 1) * 32);  // overlap next tile with compute
      // own tile-kb loads done when <=12 (next tile's) remain (in-order)
      asm volatile("s_wait_asynccnt 0xc" ::: "memory");
    } else {
      asm volatile("s_wait_asynccnt 0x0" ::: "memory");
    }
    __syncthreads();  // tile kb visible from every wave's async writes

    const float* xb = &sx[p][0];
    const float* qb = &sq[p][0];
#pragma unroll
    for (int kk = 0; kk < 32; kk += 4) {
      v2f afrag[4], bfrag[2];
      // A fragment (16x4): lane l holds A[l%16][2*(l/16)+j], j=0,1
#pragma unroll
      for (int tm = 0; tm < 4; ++tm) {
        const float* ap =
            &xb[(wm * 64 + tm * 16 + lmod) * kST + kk + 2 * lhalf];
        afrag[tm][0] = ap[0];
        afrag[tm][1] = ap[1];
      }
      // B fragment (4x16): lane l holds B[2*(l/16)+j][l%16] = Qt[n][k..k+1]
#pragma unroll
      for (int tn = 0; tn < 2; ++tn) {
        const float* bp =
            &qb[(wn * 32 + tn * 16 + lmod) * kST + kk + 2 * lhalf];
        bfrag[tn][0] = bp[0];
        bfrag[tn][1] = bp[1];
      }
#pragma unroll
      for (int tm = 0; tm < 4; ++tm)
#pragma unroll
        for (int tn = 0; tn < 2; ++tn)
          acc[tm][tn] = __builtin_amdgcn_wmma_f32_16x16x4_f32(
              false, afrag[tm], false, bfrag[tn], (short)0, acc[tm][tn],
              false, false);
    }
    __syncthreads();  // all waves done reading buf p before it is re-filled
  }

  // epilogue: D layout -> VGPR r is row (r + 8*lhalf), col lmod; add bias
#pragma unroll
  for (int tm = 0; tm < 4; ++tm) {
#pragma unroll
    for (int tn = 0; tn < 2; ++tn) {
      const int gm = mBase + wm * 64 + tm * 16 + 8 * lhalf;
      const int gn = nBase + wn * 32 + tn * 16 + lmod;
      const float bv = bias[gn];
#pragma unroll
      for (int r = 0; r < 8; ++r)
        out[(size_t)(gm + r) * kD + gn] = acc[tm][tn][r] + bv;
    }
  }
}

// ---------------------------------------------------------------------------
extern "C" void kernel_launch(void* const* d_in, const int* in_sizes, int n_in,
                              void* d_out, int out_size, void* d_ws,
                              size_t ws_size, hipStream_t stream) {
  const float* x = (const float*)d_in[0];
  const float* v = (const float*)d_in[1];
  const float* b = (const float*)d_in[2];
  float* out = (float*)d_out;

  // workspace: [s: NV][vt: NV*D][Qt: D*D]  (~4.72 MB)
  float* s = (float*)d_ws;
  float* vt = s + kNV;
  float* Qt = vt + (size_t)kNV * kD;

  uk_prep<<<kNV, 256, 0, stream>>>(v, s, vt);
  uk_house<<<kD / 32, 256, 0, stream>>>(vt, s, Qt);
  uk_gemm<<<dim3(kD / 64, kB / 128), 128, 0, stream>>>(x, Qt, b, out);
}